// GCN_15461882265887
// MI455X (gfx1250) — compile-verified
//
#include <hip/hip_runtime.h>

#define N_NODES 50000
#define N_EDGES 800000
#define DIN 128
#define DH  128
#define DOUT 64

typedef __attribute__((ext_vector_type(2))) float v2f;
typedef __attribute__((ext_vector_type(8))) float v8f;

// ---------------- degree / normalization ----------------
__global__ void deg_init_kernel(float* __restrict__ deg, int n) {
    int i = blockIdx.x * blockDim.x + threadIdx.x;
    if (i < n) deg[i] = 1.0f;                 // self-loop contributes 1
}

__global__ void deg_accum_kernel(const int* __restrict__ dst, float* __restrict__ deg, int e) {
    int i = blockIdx.x * blockDim.x + threadIdx.x;
    if (i < e)
        __hip_atomic_fetch_add(&deg[dst[i]], 1.0f, __ATOMIC_RELAXED, __HIP_MEMORY_SCOPE_AGENT);
}

__global__ void deg_rsqrt_kernel(float* __restrict__ deg, int n) {
    int i = blockIdx.x * blockDim.x + threadIdx.x;
    if (i < n) deg[i] = rsqrtf(deg[i]);       // deg >= 1 always
}

// ---------------- fp32 WMMA GEMM: C[M x NT*16] = A[M x K] * B[K x NT*16] ----
// One wave per 16x16 output tile; wave w of the block handles column tile w.
// grid.x = M/16 (M is a multiple of 16), block = 32*NTILES threads.
template <int K, int NTILES>
__global__ __launch_bounds__(32 * NTILES)
void gemm_wmma_kernel(const float* __restrict__ A, const float* __restrict__ B,
                      float* __restrict__ C) {
    constexpr int NC = NTILES * 16;
    const int wave = threadIdx.x >> 5;
    const int lane = threadIdx.x & 31;
    const int half = lane >> 4;      // 0: lanes 0-15, 1: lanes 16-31
    const int row  = lane & 15;
    const int m0 = blockIdx.x * 16;
    const int n0 = wave * 16;

    // A 16x4 f32 layout: lane holds row M=row; VGPR0/1 = K = 2*half, 2*half+1
    const float* Ap = A + (size_t)(m0 + row) * K + 2 * half;
    // B 4x16 f32 layout: lane holds col N=row; VGPR0/1 = K = 2*half, 2*half+1
    const float* Bp = B + (size_t)(2 * half) * NC + n0 + row;

    v8f c = {};
#pragma unroll 4
    for (int k = 0; k < K; k += 4) {
        v2f a = *(const v2f*)(Ap + k);            // 8B-aligned: k, 2*half even
        v2f b;
        b.x = Bp[(size_t)k * NC];
        b.y = Bp[(size_t)(k + 1) * NC];
        c = __builtin_amdgcn_wmma_f32_16x16x4_f32(
                /*neg_a=*/false, a, /*neg_b=*/false, b,
                /*c_mod=*/(short)0, c, /*reuse_a=*/false, /*reuse_b=*/false);
    }

    // C/D 16x16 f32 layout: VGPR i -> M = i + 8*half, N = row
    float* Cp = C + (size_t)(m0 + half * 8) * NC + n0 + row;
#pragma unroll
    for (int i = 0; i < 8; ++i)
        Cp[(size_t)i * NC] = c[i];
}

// ---------------- edge scatter: agg[dst] += t[src] * dinv[src]*dinv[dst] ----
// One wave per edge, lane-strided features (coalesced loads & atomics; all
// feature rows resident in the 192MB L2 so atomics resolve at L2).
template <int D>
__global__ __launch_bounds__(256)
void scatter_kernel(const float* __restrict__ t, const int* __restrict__ src,
                    const int* __restrict__ dst, const float* __restrict__ dinv,
                    float* __restrict__ agg, int e) {
    int edge = (int)((blockIdx.x * blockDim.x + threadIdx.x) >> 5);
    int lane = threadIdx.x & 31;
    if (edge >= e) return;
    int s = src[edge];
    int d = dst[edge];
    float nrm = dinv[s] * dinv[d];
    const float* ts = t + (size_t)s * D;
    float* ad = agg + (size_t)d * D;
#pragma unroll
    for (int i = 0; i < D / 32; ++i) {
        float v = ts[lane + i * 32] * nrm;
        __hip_atomic_fetch_add(&ad[lane + i * 32], v,
                               __ATOMIC_RELAXED, __HIP_MEMORY_SCOPE_AGENT);
    }
}

// ---------------- finalize: out = agg + t*dinv^2 (self loop) + bias [, relu]
template <int D, bool RELU>
__global__ void finalize_kernel(const float* __restrict__ t, const float* __restrict__ dinv,
                                const float* __restrict__ bias, float* __restrict__ agg) {
    size_t idx = (size_t)blockIdx.x * blockDim.x + threadIdx.x;
    if (idx >= (size_t)N_NODES * D) return;
    int node = (int)(idx / D);
    int f    = (int)(idx % D);
    float di = dinv[node];
    float v = agg[idx] + t[idx] * (di * di) + bias[f];
    if (RELU) v = fmaxf(v, 0.0f);
    agg[idx] = v;
}

extern "C" void kernel_launch(void* const* d_in, const int* in_sizes, int n_in,
                              void* d_out, int out_size, void* d_ws, size_t ws_size,
                              hipStream_t stream) {
    const float* x  = (const float*)d_in[0];
    const int*   ei = (const int*)  d_in[1];   // [2, E]: row0=src, row1=dst
    const float* W1 = (const float*)d_in[2];
    const float* b1 = (const float*)d_in[3];
    const float* W2 = (const float*)d_in[4];
    const float* b2 = (const float*)d_in[5];
    float* out = (float*)d_out;

    const int* src = ei;
    const int* dst = ei + N_EDGES;

    float* ws   = (float*)d_ws;
    float* dinv = ws;                                   // N floats
    float* t1   = ws + ((N_NODES + 255) & ~255);        // N*DH
    float* agg1 = t1 + (size_t)N_NODES * DH;            // N*DH (becomes H1)
    float* t2   = agg1 + (size_t)N_NODES * DH;          // N*DOUT

    // ---- symmetric normalization: dinv = rsqrt(1 + in-degree) ----
    deg_init_kernel <<<(N_NODES + 255) / 256, 256, 0, stream>>>(dinv, N_NODES);
    deg_accum_kernel<<<(N_EDGES + 255) / 256, 256, 0, stream>>>(dst, dinv, N_EDGES);
    deg_rsqrt_kernel<<<(N_NODES + 255) / 256, 256, 0, stream>>>(dinv, N_NODES);

    // ---- layer 1: T1 = X @ W1 ; AGG1 = scatter ; H1 = relu(AGG1 + self + b1)
    gemm_wmma_kernel<DIN, DH / 16><<<N_NODES / 16, 32 * (DH / 16), 0, stream>>>(x, W1, t1);
    (void)hipMemsetAsync(agg1, 0, (size_t)N_NODES * DH * sizeof(float), stream);
    scatter_kernel<DH><<<N_EDGES / 8, 256, 0, stream>>>(t1, src, dst, dinv, agg1, N_EDGES);
    finalize_kernel<DH, true><<<((size_t)N_NODES * DH + 255) / 256, 256, 0, stream>>>(t1, dinv, b1, agg1);

    // ---- layer 2: T2 = H1 @ W2 ; out = scatter + self + b2 ----
    gemm_wmma_kernel<DH, DOUT / 16><<<N_NODES / 16, 32 * (DOUT / 16), 0, stream>>>(agg1, W2, t2);
    (void)hipMemsetAsync(out, 0, (size_t)N_NODES * DOUT * sizeof(float), stream);
    scatter_kernel<DOUT><<<N_EDGES / 8, 256, 0, stream>>>(t2, src, dst, dinv, out, N_EDGES);
    finalize_kernel<DOUT, false><<<((size_t)N_NODES * DOUT + 255) / 256, 256, 0, stream>>>(t2, dinv, b2, out);
}